// MovingWindowHyperbolicForecaster_39341900431912
// MI455X (gfx1250) — compile-verified
//
#include <hip/hip_runtime.h>

// MovingWindowHyperbolicForecaster on MI455X (gfx1250, wave32).
// Memory-bound (~507MB streamed once -> ~22us floor @ 23.3 TB/s).
// One wave per batch element; 128 spatial dims = 4 floats/lane.
// Bulk data streamed via gfx1250 async global->LDS copies (ASYNCcnt),
// double-buffered so vector n+1 is in flight while logmap(n-1,n) computes.
// Streaming loads use TH=NT (data is touched once; 507MB > 192MB L2).

#define NSTEP      15
#define DP1        129
#define MEAN_ITERS 10
#define WPB        8      // waves per block (256 threads)
#define EPSF       1e-6f
#define MAX_TN     10.0f
#define CPOL_NT    1      // load temporal hint: non-temporal (ISA 7.3)

// ---- address-space helpers (integer round trip; as(3) ptr == LDS offset) ---
typedef __attribute__((address_space(1))) int as1_int;
typedef __attribute__((address_space(3))) int as3_int;

__device__ __forceinline__ as1_int* to_global(const void* p) {
  return (as1_int*)(unsigned long long)(p);
}
__device__ __forceinline__ as3_int* to_lds(const void* p) {
  // generic LDS address: low 32 bits are the LDS byte offset
  return (as3_int*)(unsigned int)(unsigned long long)(p);
}

#if defined(__has_builtin)
#if __has_builtin(__builtin_amdgcn_global_load_async_to_lds_b32)
#define HAVE_ASYNC_BUILTIN 1
#endif
#endif

__device__ __forceinline__ void async_cp_b32(const float* g, const float* l) {
#ifdef HAVE_ASYNC_BUILTIN
  __builtin_amdgcn_global_load_async_to_lds_b32(to_global(g), to_lds(l), 0, CPOL_NT);
#else
  unsigned int loff = (unsigned int)(unsigned long long)l;
  asm volatile("global_load_async_to_lds_b32 %0, %1, off th:TH_LOAD_NT"
               :: "v"(loff), "v"(g) : "memory");
#endif
}

#define S_WAIT_ASYNC_4() asm volatile("s_wait_asynccnt 4" ::: "memory")
#define S_WAIT_ASYNC_0() asm volatile("s_wait_asynccnt 0" ::: "memory")
#define S_WAIT_DS_0()    asm volatile("s_wait_dscnt 0"    ::: "memory")

// ---- wave32 reduction -------------------------------------------------------
__device__ __forceinline__ float wave_sum(float v) {
  v += __shfl_xor(v, 1);
  v += __shfl_xor(v, 2);
  v += __shfl_xor(v, 4);
  v += __shfl_xor(v, 8);
  v += __shfl_xor(v, 16);
  return v;
}

// ---- hyperbolic vector: 4 spatial comps/lane + (wave-uniform) time ----------
struct HVec { float t, x0, x1, x2, x3; };

__device__ __forceinline__ float linner(const HVec& u, const HVec& v) {
  float s = u.x0*v.x0 + u.x1*v.x1 + u.x2*v.x2 + u.x3*v.x3;
  return wave_sum(s) - u.t * v.t;
}

__device__ __forceinline__ HVec logmap(const HVec& x, const HVec& y) {
  float alpha = fmaxf(-linner(x, y), 1.0f + EPSF);
  float q     = alpha * alpha - 1.0f;
  float denom = sqrtf(q);
  float c     = logf(alpha + denom) / denom;   // arccosh(alpha)/denom
  HVec r;
  r.t  = c * (y.t  - alpha * x.t);
  r.x0 = c * (y.x0 - alpha * x.x0);
  r.x1 = c * (y.x1 - alpha * x.x1);
  r.x2 = c * (y.x2 - alpha * x.x2);
  r.x3 = c * (y.x3 - alpha * x.x3);
  return r;
}

__device__ __forceinline__ HVec logmap0(const HVec& y) {  // logmap(origin, y)
  float alpha = fmaxf(y.t, 1.0f + EPSF);
  float q     = alpha * alpha - 1.0f;
  float denom = sqrtf(q);
  float c     = logf(alpha + denom) / denom;
  HVec r;
  r.t  = c * (y.t - alpha);
  r.x0 = c * y.x0;  r.x1 = c * y.x1;  r.x2 = c * y.x2;  r.x3 = c * y.x3;
  return r;
}

__device__ __forceinline__ HVec safe_expmap(const HVec& x, const HVec& u) {
  float n2 = fmaxf(linner(u, u), EPSF);
  float nr = sqrtf(n2);
  float n  = fminf(nr, MAX_TN);
  float ch = coshf(n);
  float cs = sinhf(n) / nr;
  HVec r;
  r.t  = ch * x.t  + cs * u.t;
  r.x0 = ch * x.x0 + cs * u.x0;
  r.x1 = ch * x.x1 + cs * u.x1;
  r.x2 = ch * x.x2 + cs * u.x2;
  r.x3 = ch * x.x3 + cs * u.x3;
  return r;
}

__device__ __forceinline__ HVec safe_expmap0(const HVec& u) {  // x = origin
  float n2 = fmaxf(linner(u, u), EPSF);
  float nr = sqrtf(n2);
  float n  = fminf(nr, MAX_TN);
  float ch = coshf(n);
  float cs = sinhf(n) / nr;
  HVec r;
  r.t  = ch + cs * u.t;
  r.x0 = cs * u.x0;  r.x1 = cs * u.x1;  r.x2 = cs * u.x2;  r.x3 = cs * u.x3;
  return r;
}

__device__ __forceinline__ HVec projx(const HVec& x) {
  float s = wave_sum(x.x0*x.x0 + x.x1*x.x1 + x.x2*x.x2 + x.x3*x.x3);
  HVec r = x;
  r.t = sqrtf(1.0f + s);
  return r;
}

__device__ __forceinline__ void axpy(HVec& a, float w, const HVec& v) {
  a.t  += w * v.t;
  a.x0 += w * v.x0;  a.x1 += w * v.x1;  a.x2 += w * v.x2;  a.x3 += w * v.x3;
}

// ---- async staging: one 129-vector -> 128 spatial floats into LDS -----------
// instr k copies g[1 + 32k + lane] -> lds[lane*4 + k]  (each copy: one fully
// coalesced 128B line across the wave; lane ends up holding elems l,l+32,l+64,l+96)
__device__ __forceinline__ void issue_vec(const float* gvec, float* lbuf, int lane) {
  async_cp_b32(gvec + 1  + lane, lbuf + lane * 4 + 0);
  async_cp_b32(gvec + 33 + lane, lbuf + lane * 4 + 1);
  async_cp_b32(gvec + 65 + lane, lbuf + lane * 4 + 2);
  async_cp_b32(gvec + 97 + lane, lbuf + lane * 4 + 3);
}

__device__ __forceinline__ HVec read_vec(const float* gvec, const float* lbuf, int lane) {
  const float4 f = *(const float4*)(lbuf + lane * 4);   // ds_load_b128
  S_WAIT_DS_0();            // read fully retired before this buffer is re-filled
  HVec v;
  v.t  = __builtin_nontemporal_load(gvec);  // uniform time component, 1 dword
  v.x0 = f.x; v.x1 = f.y; v.x2 = f.z; v.x3 = f.w;
  return v;
}

__host__ __device__ constexpr float geosum14() {  // sum_{k=0..13} 0.9^k
  float s = 0.f, p = 1.f;
  for (int i = 0; i < 14; ++i) { s += p; p *= 0.9f; }
  return s;
}

__global__ __launch_bounds__(256, 1)
void hyperbolic_forecaster_kernel(const float* __restrict__ z0,
                                  const float* __restrict__ z1,
                                  const float* __restrict__ z2,
                                  const float* __restrict__ z3,
                                  const float* __restrict__ lw,
                                  float* __restrict__ out, int B) {
  __shared__ __align__(16) float lds[WPB][2][128];
  const int lane = threadIdx.x & 31;
  const int wid  = threadIdx.x >> 5;
  const int b    = blockIdx.x * WPB + wid;
  if (b >= B) return;                       // wave-uniform exit

  float* buf0 = &lds[wid][0][0];
  float* buf1 = &lds[wid][1][0];

  // softmax over the 4 lorentz weights (uniform scalars)
  float w0 = lw[0], w1 = lw[1], w2 = lw[2], w3 = lw[3];
  float mx = fmaxf(fmaxf(w0, w1), fmaxf(w2, w3));
  float e0 = expf(w0 - mx), e1 = expf(w1 - mx), e2 = expf(w2 - mx), e3 = expf(w3 - mx);
  float ei = 1.0f / (e0 + e1 + e2 + e3);
  const float ws[4] = { e0 * ei, e1 * ei, e2 * ei, e3 * ei };

  const float invS = 1.0f / geosum14();
  HVec p[4];

  #pragma unroll
  for (int s = 0; s < 4; ++s) {
    const float* z  = (s == 0) ? z0 : (s == 1) ? z1 : (s == 2) ? z2 : z3;
    const float* zb = z + (size_t)b * (NSTEP * DP1);

    // ---- software-pipelined stream: double-buffered async global->LDS ----
    issue_vec(zb + 0 * DP1, buf0, lane);
    issue_vec(zb + 1 * DP1, buf1, lane);
    S_WAIT_ASYNC_4();                                  // vec0 landed
    HVec prev = read_vec(zb + 0 * DP1, buf0, lane);
    HVec acc  = {0.f, 0.f, 0.f, 0.f, 0.f};

    for (int n = 1; n <= 13; ++n) {
      issue_vec(zb + (n + 1) * DP1, (n & 1) ? buf0 : buf1, lane); // vec n+1
      S_WAIT_ASYNC_4();                                // vec n landed
      HVec cur = read_vec(zb + n * DP1, (n & 1) ? buf1 : buf0, lane);
      HVec v   = logmap(prev, cur);
      // Horner: acc = 0.9*acc + v  =>  sum_i 0.9^(13-i) v_i
      acc.t  = acc.t  * 0.9f + v.t;
      acc.x0 = acc.x0 * 0.9f + v.x0;
      acc.x1 = acc.x1 * 0.9f + v.x1;
      acc.x2 = acc.x2 * 0.9f + v.x2;
      acc.x3 = acc.x3 * 0.9f + v.x3;
      prev = cur;
    }
    S_WAIT_ASYNC_0();                                  // drain: vec14 landed
    {
      HVec cur = read_vec(zb + 14 * DP1, buf0, lane);
      HVec v   = logmap(prev, cur);
      acc.t  = acc.t  * 0.9f + v.t;
      acc.x0 = acc.x0 * 0.9f + v.x0;
      acc.x1 = acc.x1 * 0.9f + v.x1;
      acc.x2 = acc.x2 * 0.9f + v.x2;
      acc.x3 = acc.x3 * 0.9f + v.x3;
      prev = cur;
    }
    acc.t *= invS; acc.x0 *= invS; acc.x1 *= invS; acc.x2 *= invS; acc.x3 *= invS;
    p[s] = projx(safe_expmap(prev, acc));              // predicted next point
  }

  // ---- weighted Lorentz (Karcher) mean over the 4 predictions ----
  HVec wt = {0.f, 0.f, 0.f, 0.f, 0.f};
  #pragma unroll
  for (int s = 0; s < 4; ++s) axpy(wt, ws[s], logmap0(p[s]));
  HVec mean = projx(safe_expmap0(wt));

  for (int it = 0; it < MEAN_ITERS; ++it) {
    HVec wv = {0.f, 0.f, 0.f, 0.f, 0.f};
    #pragma unroll
    for (int s = 0; s < 4; ++s) axpy(wv, ws[s], logmap(mean, p[s]));
    HVec h;
    h.t  = 0.5f * wv.t;
    h.x0 = 0.5f * wv.x0; h.x1 = 0.5f * wv.x1;
    h.x2 = 0.5f * wv.x2; h.x3 = 0.5f * wv.x3;
    mean = projx(safe_expmap(mean, h));
  }

  float* ob = out + (size_t)b * DP1;
  if (lane == 0) ob[0] = mean.t;
  ob[1  + lane] = mean.x0;
  ob[33 + lane] = mean.x1;
  ob[65 + lane] = mean.x2;
  ob[97 + lane] = mean.x3;
}

extern "C" void kernel_launch(void* const* d_in, const int* in_sizes, int n_in,
                              void* d_out, int out_size, void* d_ws, size_t ws_size,
                              hipStream_t stream) {
  const float* z0 = (const float*)d_in[0];
  const float* z1 = (const float*)d_in[1];
  const float* z2 = (const float*)d_in[2];
  const float* z3 = (const float*)d_in[3];
  const float* lw = (const float*)d_in[4];
  float* out = (float*)d_out;
  int B = in_sizes[0] / (NSTEP * DP1);
  int blocks = (B + WPB - 1) / WPB;
  hyperbolic_forecaster_kernel<<<blocks, 256, 0, stream>>>(z0, z1, z2, z3, lw, out, B);
}